// NodeDropout_16801912062155
// MI455X (gfx1250) — compile-verified
//
#include <hip/hip_runtime.h>

// NodeDropout edge masking:
//   out[e] = values[e] * (!nodes_flag[src[e]]) * (!nodes_flag[dst[e]])
//
// Pure HBM-bandwidth-bound streaming gather (AI ~0.125 FLOP/B). Optimized for
// MI455X (gfx1250, wave32):
//   - 128-bit vector loads/stores (4 edges per thread/lane)
//   - NON-TEMPORAL hints on the once-touched streams (edge_index, values, out)
//     so the ~1 MB nodes_flag table stays resident in the 192 MB L2 and the
//     random byte-gathers hit cache.
//   - branchless mask via select (no divergence).

typedef int   v4i __attribute__((ext_vector_type(4)));
typedef float v4f __attribute__((ext_vector_type(4)));

__global__ __launch_bounds__(256) void node_dropout_vec4(
    const int*           __restrict__ src_idx,     // edge_index row 0 (E)
    const int*           __restrict__ dst_idx,     // edge_index row 1 (E)
    const float*         __restrict__ values,      // E
    const unsigned char* __restrict__ nodes_flag,  // N, 1 => node dropped
    float*               __restrict__ out,         // E
    long long n4)                                  // number of 4-edge groups
{
    long long g = (long long)blockIdx.x * blockDim.x + threadIdx.x;
    if (g >= n4) return;
    long long e = g << 2;

    // Streaming data: touched exactly once -> non-temporal (global_load_b128 NT)
    v4i s = __builtin_nontemporal_load((const v4i*)(src_idx + e));
    v4i d = __builtin_nontemporal_load((const v4i*)(dst_idx + e));
    v4f v = __builtin_nontemporal_load((const v4f*)(values  + e));

    // Flag gathers: default (RT) temporal policy -> cached in WGP$/L2.
    v4f r;
#pragma unroll
    for (int k = 0; k < 4; ++k) {
        unsigned int drop = (unsigned int)nodes_flag[s[k]] |
                            (unsigned int)nodes_flag[d[k]];
        r[k] = drop ? 0.0f : v[k];   // branchless v_cndmask
    }

    __builtin_nontemporal_store(r, (v4f*)(out + e));
}

// Scalar tail for E % 4 != 0 (E = 20M is divisible by 4; kept for generality).
__global__ __launch_bounds__(64) void node_dropout_tail(
    const int*           __restrict__ src_idx,
    const int*           __restrict__ dst_idx,
    const float*         __restrict__ values,
    const unsigned char* __restrict__ nodes_flag,
    float*               __restrict__ out,
    long long start, long long E)
{
    long long e = start + threadIdx.x;
    if (e >= E) return;
    unsigned int drop = (unsigned int)nodes_flag[src_idx[e]] |
                        (unsigned int)nodes_flag[dst_idx[e]];
    float v = __builtin_nontemporal_load(values + e);
    __builtin_nontemporal_store(drop ? 0.0f : v, out + e);
}

extern "C" void kernel_launch(void* const* d_in, const int* in_sizes, int n_in,
                              void* d_out, int out_size, void* d_ws, size_t ws_size,
                              hipStream_t stream) {
    // Inputs per reference setup_inputs() order:
    //   d_in[0]: edge_index, shape (2, E) flattened -> int32 per harness convention
    //   d_in[1]: values, float32 (E)
    //   d_in[2]: nodes_flag, bool (N) -> 1 byte each
    const int*           edge_index = (const int*)d_in[0];
    const float*         values     = (const float*)d_in[1];
    const unsigned char* nodes_flag = (const unsigned char*)d_in[2];
    float*               out        = (float*)d_out;

    const long long E   = (long long)in_sizes[0] / 2;  // 20,000,000
    const int* src = edge_index;        // row 0
    const int* dst = edge_index + E;    // row 1

    const long long n4   = E >> 2;      // 4 edges per thread
    const long long tail = E - (n4 << 2);

    const int BLOCK = 256;              // 8 wave32s per block
    if (n4 > 0) {
        unsigned grid = (unsigned)((n4 + BLOCK - 1) / BLOCK);
        node_dropout_vec4<<<grid, BLOCK, 0, stream>>>(
            src, dst, values, nodes_flag, out, n4);
    }
    if (tail > 0) {
        node_dropout_tail<<<1, 64, 0, stream>>>(
            src, dst, values, nodes_flag, out, n4 << 2, E);
    }
}